// custom_weighted_loss_2_10514079940984
// MI455X (gfx1250) — compile-verified
//
#include <hip/hip_runtime.h>

// ---------------------------------------------------------------------------
// Weighted MSE loss, single streaming pass + tiny epilogue.
//   loss = (1/N) * sum_slice mult[slice] * sum_pixels base*(p-t)^2
//   base = (t>0.5) ? 10 : 1
//   h[b,d] = any(t>0.5) over HxW; mult = h?1 : ((prev_h|next_h)?0:5)
// HBM-bound: 436 MB read once -> ~19us floor at 23.3 TB/s.
// ---------------------------------------------------------------------------

typedef float v2f __attribute__((ext_vector_type(2)));
typedef float v4f __attribute__((ext_vector_type(4)));
typedef float v8f __attribute__((ext_vector_type(8)));

#define NB 16
#define ND 13
#define NSLICES (NB * ND)        // 208
#define HW (512 * 512)           // 262144 elements per slice
#define BPS 32                   // blocks per slice
#define NPART (NSLICES * BPS)    // 6656 block partials
#define TPB 256                  // 8 waves (wave32)
#define EPB (HW / BPS)           // 8192 elements per block
#define VPT (EPB / (TPB * 4))    // 8 float4 per thread per array

__global__ void zero_flags_kernel(int* __restrict__ hflags) {
  const int i = threadIdx.x;
  if (i < NSLICES) hflags[i] = 0;
}

__global__ __launch_bounds__(TPB) void wloss_partial_kernel(
    const float* __restrict__ pred, const float* __restrict__ targ,
    float* __restrict__ partials, int* __restrict__ hflags) {
  __shared__ float wsum[TPB / 32];
  __shared__ int sfg;
  if (threadIdx.x == 0) sfg = 0;
  __syncthreads();

  const int blk = blockIdx.x;
  const int slice = blk / BPS;
  const int sub = blk - slice * BPS;
  const size_t base = (size_t)slice * HW + (size_t)sub * EPB;
  const v4f* __restrict__ p4 = (const v4f*)(pred + base);
  const v4f* __restrict__ t4 = (const v4f*)(targ + base);

  float acc = 0.0f;
  int fgany = 0;
#pragma unroll
  for (int i = 0; i < VPT; ++i) {
    const int idx = i * TPB + threadIdx.x;
    const v4f p = __builtin_nontemporal_load(&p4[idx]);   // streaming, no reuse
    const v4f t = __builtin_nontemporal_load(&t4[idx]);
#pragma unroll
    for (int c = 0; c < 4; ++c) {
      const float tv = t[c];
      const float d = p[c] - tv;
      const bool fg = tv > 0.5f;
      acc = fmaf(d * d, fg ? 10.0f : 1.0f, acc);
      fgany |= (int)fg;
    }
  }
  if (fgany) atomicOr(&sfg, 1);   // LDS, idempotent -> deterministic

  // Cross-lane wave reduction on the matrix pipe:
  //   D(16x16) = A(16x4) x B(4x16) with B = ones  =>  D[m][n] = sum_k A[m][k].
  // Each lane contributes acc exactly once into A (second VGPR zeroed), so
  // sum over the 8 D VGPRs at lanes 0 and 16 yields the full 32-lane total,
  // independent of B's exact lane layout.
  v2f A; A.x = acc;  A.y = 0.0f;
  v2f Bm; Bm.x = 1.0f; Bm.y = 1.0f;
  v8f C = {};
  C = __builtin_amdgcn_wmma_f32_16x16x4_f32(
      /*neg_a=*/false, A, /*neg_b=*/false, Bm,
      /*c_mod=*/(short)0, C, /*reuse_a=*/false, /*reuse_b=*/false);
  const float s =
      C[0] + C[1] + C[2] + C[3] + C[4] + C[5] + C[6] + C[7];
  const float wtot = __shfl(s, 0, 32) + __shfl(s, 16, 32);

  const int wave = threadIdx.x >> 5;
  const int lane = threadIdx.x & 31;
  if (lane == 0) wsum[wave] = wtot;
  __syncthreads();

  if (threadIdx.x == 0) {
    float tot = 0.0f;
#pragma unroll
    for (int w = 0; w < TPB / 32; ++w) tot += wsum[w];
    partials[blk] = tot;                       // unique slot -> deterministic
    if (sfg) atomicOr(&hflags[slice], 1);      // idempotent OR
  }
}

__global__ __launch_bounds__(TPB) void wloss_finalize_kernel(
    const float* __restrict__ partials, const int* __restrict__ hflags,
    float* __restrict__ out) {
  __shared__ double wS[NSLICES];
  const int s = threadIdx.x;
  if (s < NSLICES) {
    double acc = 0.0;
    for (int k = 0; k < BPS; ++k) acc += (double)partials[s * BPS + k];
    const int d = s % ND;
    const int h = hflags[s];
    const int ph = (d > 0) ? hflags[s - 1] : 0;
    const int nh = (d < ND - 1) ? hflags[s + 1] : 0;
    const double mult = h ? 1.0 : ((ph | nh) ? 0.0 : 5.0);
    wS[s] = acc * mult;
  }
  __syncthreads();
  if (s == 0) {
    double tot = 0.0;
    for (int i = 0; i < NSLICES; ++i) tot += wS[i];   // fixed order
    out[0] = (float)(tot / (double)((size_t)NB * ND * HW));
  }
}

extern "C" void kernel_launch(void* const* d_in, const int* in_sizes, int n_in,
                              void* d_out, int out_size, void* d_ws,
                              size_t ws_size, hipStream_t stream) {
  const float* pred = (const float*)d_in[0];
  const float* targ = (const float*)d_in[1];
  float* out = (float*)d_out;

  float* partials = (float*)d_ws;                          // NPART floats
  int* hflags = (int*)((char*)d_ws + NPART * sizeof(float));  // NSLICES ints

  zero_flags_kernel<<<1, TPB, 0, stream>>>(hflags);
  wloss_partial_kernel<<<NPART, TPB, 0, stream>>>(pred, targ, partials, hflags);
  wloss_finalize_kernel<<<1, TPB, 0, stream>>>(partials, hflags, out);
}